// CIN_31198642438696
// MI455X (gfx1250) — compile-verified
//
#include <hip/hip_runtime.h>

typedef __attribute__((ext_vector_type(2))) float v2f;
typedef __attribute__((ext_vector_type(8))) float v8f;

#define FIELD 32
#define DIM   64
#define C0    1024   // 32*32
#define C1    2048   // 64*32
#define HID   64
#define OUTC  192

#if __has_builtin(__builtin_amdgcn_sched_barrier)
#define SCHED_FENCE() __builtin_amdgcn_sched_barrier(0)
#else
#define SCHED_FENCE()
#endif

__global__ __launch_bounds__(256, 2)
void cin_wmma_kernel(const float* __restrict__ x,
                     const float* __restrict__ w0, const float* __restrict__ b0,
                     const float* __restrict__ w1, const float* __restrict__ b1,
                     float* __restrict__ out)
{
    __shared__ float xl[FIELD * DIM];   // x[b]   : 32x64 f32 = 8 KB
    __shared__ float hl[HID * DIM];     // hidden : 64x64 f32 = 16 KB

    const int b    = blockIdx.x;
    const int tid  = threadIdx.x;
    const int wave = tid >> 5;          // 0..7 (wave32)
    const int lane = tid & 31;
    const int ln   = lane & 15;         // tile col (N) / tile row (M) index
    const int hi   = lane >> 4;         // lane half
    const int kb   = hi * 2;            // K sub-offset inside 4-wide K step

    // ---- stage x[b] into LDS (coalesced float4) ----
    {
        const float4* src = (const float4*)(x + (size_t)b * (FIELD * DIM));
        float4* dst = (float4*)xl;
        #pragma unroll
        for (int i = tid; i < (FIELD * DIM) / 4; i += 256) dst[i] = src[i];
    }
    __syncthreads();

    const int m0 = wave * 16;           // this wave's M-tile of the 128 output channels

    // ---- materialize the f-side outer-product values in VGPRs once ----
    // Lane needs x[f, d] for f = 4q + kb + i (16 rows) x 4 d-tiles.
    // These are identical for BOTH layers (f-side is always x).
    float xf[16][4];
    #pragma unroll
    for (int q = 0; q < 8; ++q)
        #pragma unroll
        for (int i = 0; i < 2; ++i)
            #pragma unroll
            for (int t = 0; t < 4; ++t)
                xf[2 * q + i][t] = xl[(4 * q + kb + i) * DIM + t * 16 + ln];

    // One h-step = 8 A fragments (W row cols h*32..h*32+31) + 4 h-side values,
    // feeding 32 WMMAs. Double-buffered so loads stay >= one compute phase ahead.
    auto load_step = [&](v2f af[8], float xh[4], const float* wlayer, int cstride,
                         const float* hsrc, int h) {
        const float* wh = wlayer + (size_t)(m0 + ln) * cstride + h * FIELD + kb;
        #pragma unroll
        for (int q = 0; q < 8; ++q) af[q] = *(const v2f*)(wh + 4 * q);
        #pragma unroll
        for (int t = 0; t < 4; ++t) xh[t] = hsrc[h * DIM + t * 16 + ln];
    };
    auto compute_step = [&](const v2f af[8], const float xh[4], v8f accp[4]) {
        #pragma unroll
        for (int q = 0; q < 8; ++q)
            #pragma unroll
            for (int t = 0; t < 4; ++t) {
                v2f bf;                               // z[k][d] = hsrc[h,d]*x[f,d]
                bf.x = xh[t] * xf[2 * q + 0][t];
                bf.y = xh[t] * xf[2 * q + 1][t];
                accp[t] = __builtin_amdgcn_wmma_f32_16x16x4_f32(
                    false, af[q], false, bf, (short)0, accp[t], false, false);
            }
    };
    auto run_layer = [&](const float* wlayer, int cstride, const float* hsrc,
                         int nh, v8f accp[4]) {
        v2f  afA[8], afB[8];
        float xhA[4], xhB[4];
        load_step(afA, xhA, wlayer, cstride, hsrc, 0);
        for (int h = 0; h < nh; h += 2) {
            load_step(afB, xhB, wlayer, cstride, hsrc, h + 1);
            SCHED_FENCE();                      // keep B-loads ahead of A-compute
            compute_step(afA, xhA, accp);
            const int hn = (h + 2 < nh) ? h + 2 : nh - 1;   // clamp: harmless reload on last iter
            load_step(afA, xhA, wlayer, cstride, hsrc, hn);
            SCHED_FENCE();                      // keep A-loads ahead of B-compute
            compute_step(afB, xhB, accp);
        }
    };

    // =========================== Layer 0 ===========================
    // y0[o,d] = relu( sum_{h,f} w0[o, h*32+f] * x[h,d]*x[f,d] + b0[o] )
    v8f acc[4];
    {
        const float* bb = b0 + m0 + 8 * hi;   // C/D layout: VGPR r -> row m0 + 8*hi + r
        v8f binit;
        #pragma unroll
        for (int r = 0; r < 8; ++r) binit[r] = bb[r];
        #pragma unroll
        for (int t = 0; t < 4; ++t) acc[t] = binit;
    }
    run_layer(w0, C0, xl, FIELD, acc);

    // ReLU
    #pragma unroll
    for (int t = 0; t < 4; ++t)
        #pragma unroll
        for (int r = 0; r < 8; ++r)
            acc[t][r] = fmaxf(acc[t][r], 0.0f);

    if (wave < 4) {
        // rows 0..63 -> hidden in LDS
        #pragma unroll
        for (int t = 0; t < 4; ++t)
            #pragma unroll
            for (int r = 0; r < 8; ++r)
                hl[(m0 + 8 * hi + r) * DIM + t * 16 + ln] = acc[t][r];
    } else {
        // rows 64..127 -> direct output: out[b, m-64] = sum_d y0[m,d]
        #pragma unroll
        for (int r = 0; r < 8; ++r) {
            float s = acc[0][r] + acc[1][r] + acc[2][r] + acc[3][r];
            s += __shfl_xor(s, 1);
            s += __shfl_xor(s, 2);
            s += __shfl_xor(s, 4);
            s += __shfl_xor(s, 8);      // butterfly stays inside 16-lane half
            if (ln == 0)
                out[(size_t)b * OUTC + (m0 - 64) + 8 * hi + r] = s;
        }
    }
    __syncthreads();

    // =========================== Layer 1 ===========================
    // y1[o,d] = relu( sum_{h<64,f<32} w1[o, h*32+f] * hidden[h,d]*x[f,d] + b1[o] )
    v8f acc1[4];
    {
        const float* bb = b1 + m0 + 8 * hi;
        v8f binit;
        #pragma unroll
        for (int r = 0; r < 8; ++r) binit[r] = bb[r];
        #pragma unroll
        for (int t = 0; t < 4; ++t) acc1[t] = binit;
    }
    run_layer(w1, C1, hl, HID, acc1);

    // ReLU + sum over d -> out[b, 64 + m]
    #pragma unroll
    for (int r = 0; r < 8; ++r) {
        float s = 0.0f;
        #pragma unroll
        for (int t = 0; t < 4; ++t) s += fmaxf(acc1[t][r], 0.0f);
        s += __shfl_xor(s, 1);
        s += __shfl_xor(s, 2);
        s += __shfl_xor(s, 4);
        s += __shfl_xor(s, 8);
        if (ln == 0)
            out[(size_t)b * OUTC + 64 + m0 + 8 * hi + r] = s;
    }
}

extern "C" void kernel_launch(void* const* d_in, const int* in_sizes, int n_in,
                              void* d_out, int out_size, void* d_ws, size_t ws_size,
                              hipStream_t stream) {
    const float* x  = (const float*)d_in[0];
    const float* w0 = (const float*)d_in[1];
    const float* b0 = (const float*)d_in[2];
    const float* w1 = (const float*)d_in[3];
    const float* b1 = (const float*)d_in[4];
    float* out = (float*)d_out;
    const int B = in_sizes[0] / (FIELD * DIM);   // 4096
    cin_wmma_kernel<<<B, 256, 0, stream>>>(x, w0, b0, w1, b1, out);
}